// NeuralConstraintProjector_67714454389373
// MI455X (gfx1250) — compile-verified
//
#include <hip/hip_runtime.h>
#include <hip/hip_bf16.h>

typedef __attribute__((ext_vector_type(2))) float v2f;
typedef __attribute__((ext_vector_type(8))) float v8f;

#define NATOMS 4096
#define TILE   16
#define WPB    8                                  // waves per block (256 thr, wave32)
#define BLOCKS_PER_BATCH ((NATOMS / TILE) / WPB)  // 32

// One wave owns a 16-row i-tile; d2 tiles come off the matrix pipe:
//   A = [-2x, -2y | -2z, 1]  (16x4, K-split across lane halves, loop-invariant)
//   B = [ xj,  yj |  zj, nj] (4x16, pre-swizzled in LDS as float2 pairs)
//   C = ni[v]                (loop-invariant)
//   D[m,n] = -2 xi.xj + nj + ni = ||xi - xj||^2
__global__ __launch_bounds__(256, 1) void steric_adjust_wmma(
    const float* __restrict__ coords,
    const float* __restrict__ radii,
    float* __restrict__ out)
{
    __shared__ float4 satom[NATOMS];   // {x, y, z, |x|^2}
    __shared__ float  srad[NATOMS];

    const int tid  = threadIdx.x;
    const int lane = tid & 31;
    const int wave = tid >> 5;
    const int b      = blockIdx.x / BLOCKS_PER_BATCH;
    const int i_base = ((blockIdx.x % BLOCKS_PER_BATCH) * WPB + wave) * TILE;

    const float* __restrict__ cb = coords + (size_t)b * NATOMS * 3;
    const float* __restrict__ rb = radii  + (size_t)b * NATOMS;

    // ---- stage whole batch into LDS (16 uniform iterations, no divergence)
    for (int a = tid; a < NATOMS; a += 256) {
        const float x = cb[3*a+0], y = cb[3*a+1], z = cb[3*a+2];
        satom[a] = make_float4(x, y, z, fmaf(z, z, fmaf(y, y, x * x)));
        srad[a]  = rb[a];
    }
    __syncthreads();

    const int half = lane >> 4;   // K-group select (A/B), row-half select (C/D)
    const int col  = lane & 15;   // n / matrix row within half

    // ---- A operand (loop-invariant): row m = i_base+col
    v2f A;
    {
        const float4 p = satom[i_base + col];
        A.x = half ? (-2.0f * p.z) : (-2.0f * p.x);
        A.y = half ? 1.0f          : (-2.0f * p.y);
    }

    // ---- epilogue rows M = v + 8*half; C = ni (loop-invariant)
    float xi[8][3], sri[8];
    v8f Cm;
#pragma unroll
    for (int v = 0; v < 8; ++v) {
        const int row  = i_base + v + 8 * half;
        const float4 p = satom[row];
        xi[v][0] = p.x; xi[v][1] = p.y; xi[v][2] = p.z;
        Cm[v]  = p.w;
        sri[v] = srad[row];
    }

    float acc[8][3];
#pragma unroll
    for (int v = 0; v < 8; ++v) {
        acc[v][0] = 0.0f; acc[v][1] = 0.0f; acc[v][2] = 0.0f;
    }

    // float2 view: atom j occupies pairs 2j (x,y) and 2j+1 (z,nj)
    const float2* __restrict__ bops = (const float2*)satom;

#pragma unroll 2
    for (int j_base = 0; j_base < NATOMS; j_base += TILE) {
        const int j = j_base + col;

        const float2 bv  = bops[2 * j + half];   // WMMA B operand: ds_load_b64,
        const float  srj = srad[j];              //   conflict-free across halves
        const float2 blo = bops[2 * j + 0];      // (x, y)   -- epilogue needs all
        const float2 bhi = bops[2 * j + 1];      // (z, nj)     three components

        v2f Bop; Bop.x = bv.x; Bop.y = bv.y;

        v8f D = __builtin_amdgcn_wmma_f32_16x16x4_f32(
            /*neg_a=*/false, A, /*neg_b=*/false, Bop,
            /*c_mod=*/(short)0, Cm, /*reuse_a=*/false, /*reuse_b=*/false);

        const float xj0 = blo.x, xj1 = blo.y, xj2 = bhi.x;
#pragma unroll
        for (int v = 0; v < 8; ++v) {
            // |d2| is a free source modifier on v_sqrt_f32; rounding can only
            // make d2 negative when true distance ~ 0, where direction ~ 0 too.
            const float dist = __builtin_amdgcn_sqrtf(__builtin_fabsf(D[v]));
            const float tgt  = fmaxf(1.0f, sri[v] + srj);
            const float pen  = fmaxf(0.0f, tgt - dist);
            const float d0 = xi[v][0] - xj0;
            const float d1 = xi[v][1] - xj1;
            const float d2c = xi[v][2] - xj2;
            acc[v][0] = fmaf(pen, fminf(1.0f, fmaxf(-1.0f, d0)),  acc[v][0]);
            acc[v][1] = fmaf(pen, fminf(1.0f, fmaxf(-1.0f, d1)),  acc[v][1]);
            acc[v][2] = fmaf(pen, fminf(1.0f, fmaxf(-1.0f, d2c)), acc[v][2]);
        }
    }

    // ---- sum over the 16 columns held within each lane-half
#pragma unroll
    for (int v = 0; v < 8; ++v)
#pragma unroll
        for (int c = 0; c < 3; ++c) {
            float s = acc[v][c];
            s += __shfl_xor(s, 1, 16);
            s += __shfl_xor(s, 2, 16);
            s += __shfl_xor(s, 4, 16);
            s += __shfl_xor(s, 8, 16);
            acc[v][c] = s;
        }

    // lanes 0 and 16 hold full row sums for M = v (+8 for upper half)
    if (col == 0) {
        const float inv_n = 1.0f / (float)NATOMS;
#pragma unroll
        for (int v = 0; v < 8; ++v) {
            const int row = i_base + v + 8 * half;
#pragma unroll
            for (int c = 0; c < 3; ++c)
                out[((size_t)b * NATOMS + row) * 3 + c] =
                    xi[v][c] + 0.1f * acc[v][c] * inv_n;
        }
    }
}

extern "C" void kernel_launch(void* const* d_in, const int* in_sizes, int n_in,
                              void* d_out, int out_size, void* d_ws, size_t ws_size,
                              hipStream_t stream) {
    const float* coords = (const float*)d_in[0];   // [B, N, 3] fp32
    const float* radii  = (const float*)d_in[1];   // [B, N]    fp32
    float* out = (float*)d_out;                    // [B, N, 3] fp32

    const int B = in_sizes[1] / NATOMS;            // 4
    const int blocks = B * BLOCKS_PER_BATCH;       // 128 blocks of 256 threads

    steric_adjust_wmma<<<blocks, 256, 0, stream>>>(coords, radii, out);
}